// IoULoss_10900626997841
// MI455X (gfx1250) — compile-verified
//
#include <hip/hip_runtime.h>
#include <hip/hip_bf16.h>

// IoU loss, C=4 classes. bin = label*4 + pred  ->  16-bin histogram.
// Stage 1: streaming count (HBM-bound, ~128 MiB @ 23.3 TB/s => ~5.8us floor)
// Stage 2: single-wave finalize using V_WMMA_F32_16X16X4_F32 for exact
//          row/col sums of the 4x4 confusion matrix.

typedef float v2f __attribute__((ext_vector_type(2)));
typedef float v8f __attribute__((ext_vector_type(8)));

__global__ __launch_bounds__(256) void iou_count_kernel(
    const float4* __restrict__ p4, const int4* __restrict__ l4,
    unsigned int* __restrict__ ghist, int n4)
{
    __shared__ unsigned int sh[16];
    if (threadIdx.x < 16) sh[threadIdx.x] = 0u;
    __syncthreads();

    unsigned int cnt[16];
#pragma unroll
    for (int i = 0; i < 16; ++i) cnt[i] = 0u;

    // Two 64-bit packed accumulators: 8 bits per bin, bins 0-7 in acc0,
    // bins 8-15 in acc1. Flush every 252 elements (max byte value 252 < 256).
    unsigned long long acc0 = 0ull, acc1 = 0ull;
    int pending = 0;

    const int stride = (int)(gridDim.x * blockDim.x);
    for (int i = (int)(blockIdx.x * blockDim.x + threadIdx.x); i < n4; i += stride) {
        __builtin_prefetch(&p4[i + 8 * stride], 0, 1);   // global_prefetch_b8
        __builtin_prefetch(&l4[i + 8 * stride], 0, 1);
        float4 p = p4[i];   // global_load_b128
        int4   l = l4[i];   // global_load_b128

        int b0 = (l.x << 2) | (int)p.x;   // preds are integer-valued floats
        int b1 = (l.y << 2) | (int)p.y;
        int b2 = (l.z << 2) | (int)p.z;
        int b3 = (l.w << 2) | (int)p.w;

        { unsigned long long o = 1ull << ((b0 & 7) << 3);
          acc0 += (b0 < 8) ? o : 0ull;  acc1 += (b0 < 8) ? 0ull : o; }
        { unsigned long long o = 1ull << ((b1 & 7) << 3);
          acc0 += (b1 < 8) ? o : 0ull;  acc1 += (b1 < 8) ? 0ull : o; }
        { unsigned long long o = 1ull << ((b2 & 7) << 3);
          acc0 += (b2 < 8) ? o : 0ull;  acc1 += (b2 < 8) ? 0ull : o; }
        { unsigned long long o = 1ull << ((b3 & 7) << 3);
          acc0 += (b3 < 8) ? o : 0ull;  acc1 += (b3 < 8) ? 0ull : o; }

        pending += 4;
        if (pending == 252) {
#pragma unroll
            for (int j = 0; j < 8; ++j) {
                cnt[j]     += (unsigned int)((acc0 >> (8 * j)) & 0xFFull);
                cnt[8 + j] += (unsigned int)((acc1 >> (8 * j)) & 0xFFull);
            }
            acc0 = 0ull; acc1 = 0ull; pending = 0;
        }
    }
    // Final flush
#pragma unroll
    for (int j = 0; j < 8; ++j) {
        cnt[j]     += (unsigned int)((acc0 >> (8 * j)) & 0xFFull);
        cnt[8 + j] += (unsigned int)((acc1 >> (8 * j)) & 0xFFull);
    }

    // Block-level reduction via LDS atomics, then 16 global atomics per block.
#pragma unroll
    for (int i = 0; i < 16; ++i) atomicAdd(&sh[i], cnt[i]);
    __syncthreads();
    if (threadIdx.x < 16) atomicAdd(&ghist[threadIdx.x], sh[threadIdx.x]);
}

// Handles the (n % 4) tail, if any (none for the 2^24 reference shape).
__global__ void iou_tail_kernel(const float* __restrict__ preds,
                                const int* __restrict__ labels,
                                unsigned int* __restrict__ ghist,
                                int start, int n)
{
    int i = start + (int)threadIdx.x;
    if (i < n) {
        int b = (labels[i] << 2) | (int)preds[i];
        atomicAdd(&ghist[b], 1u);
    }
}

// One wave32. EXEC is all-ones through both WMMAs (no divergence before them).
// WMMA #1: D1 = H(16x4, zero-padded) x ones(4x16) -> D1[m][*] = rowsum[m] (label counts)
// WMMA #2: D2 = ones(16x4) x H(4x16, zero-padded) -> D2[*][n] = colsum[n] (pred counts)
// Row/col sums are invariant to the K-slot permutation of the A/B layouts.
// Counts <= 2^24 are exact in f32, and each WMMA sums only 4 terms -> exact.
__global__ __launch_bounds__(32) void iou_finalize_kernel(
    const unsigned int* __restrict__ hist, float* __restrict__ out)
{
    const int lane = (int)threadIdx.x;      // 0..31
    const int m  = lane & 15;               // matrix row / column index
    const int kh = lane >> 4;               // which K half this lane holds

    v2f ones; ones.x = 1.0f; ones.y = 1.0f;
    v8f czero = {};

    // A1[m][k] = hist[m*4 + k] for m<4, else 0.   (rows = labels)
    v2f a1;
    a1.x = (m < 4) ? (float)hist[m * 4 + (2 * kh + 0)] : 0.0f;
    a1.y = (m < 4) ? (float)hist[m * 4 + (2 * kh + 1)] : 0.0f;
    v8f rowD = __builtin_amdgcn_wmma_f32_16x16x4_f32(
        false, a1, false, ones, (short)0, czero, false, false);

    // B2[k][n] = hist[k*4 + n] for n<4, else 0.   (cols = preds)
    v2f b2;
    b2.x = (m < 4) ? (float)hist[(2 * kh + 0) * 4 + m] : 0.0f;
    b2.y = (m < 4) ? (float)hist[(2 * kh + 1) * 4 + m] : 0.0f;
    v8f colD = __builtin_amdgcn_wmma_f32_16x16x4_f32(
        false, ones, false, b2, (short)0, czero, false, false);

    // C/D layout: lanes 0-15 hold column n=lane; VGPR j = row M=j.
    __shared__ float rows[4], cols[4];
    if (lane == 0) {                 // lane 0 holds rows 0..7 in components 0..7
        rows[0] = rowD[0]; rows[1] = rowD[1];
        rows[2] = rowD[2]; rows[3] = rowD[3];
    }
    if (lane < 4) cols[lane] = colD[0];   // component 0 = D[0][lane] = colsum[lane]
    __syncthreads();

    if (lane < 4) {
        float inter = (float)hist[lane * 4 + lane];
        float uni   = rows[lane] + cols[lane] - inter;
        float iou   = (uni == 0.0f) ? 1.0f : (inter / uni);
        out[lane]   = 1.0f - 100.0f * iou;
    }
}

extern "C" void kernel_launch(void* const* d_in, const int* in_sizes, int n_in,
                              void* d_out, int out_size, void* d_ws, size_t ws_size,
                              hipStream_t stream) {
    const float* preds  = (const float*)d_in[0];
    const int*   labels = (const int*)d_in[1];
    float*       out    = (float*)d_out;
    unsigned int* ghist = (unsigned int*)d_ws;

    const int n  = in_sizes[0];          // 16*1024*1024 fits in int
    const int n4 = n >> 2;
    const int tail_start = n4 << 2;

    hipMemsetAsync(d_ws, 0, 16 * sizeof(unsigned int), stream);

    const int threads = 256;             // 8 waves per block on wave32
    int blocks = 2048;
    if ((long long)blocks * threads > (long long)n4)
        blocks = (n4 + threads - 1) / threads;
    if (blocks < 1) blocks = 1;

    if (n4 > 0) {
        iou_count_kernel<<<blocks, threads, 0, stream>>>(
            (const float4*)preds, (const int4*)labels, ghist, n4);
    }
    if (tail_start < n) {
        iou_tail_kernel<<<1, 32, 0, stream>>>(preds, labels, ghist, tail_start, n);
    }
    iou_finalize_kernel<<<1, 32, 0, stream>>>(ghist, out);
}